// Encoder_48095043780825
// MI455X (gfx1250) — compile-verified
//
#include <hip/hip_runtime.h>
#include <hip/hip_bf16.h>

#define DIM 128
#define LDS_STRIDE 144   // DIM + 16 floats: kills 2-way LDS bank conflicts
                         // (144 mod 64 = 16 -> row k+2 is 32 banks away from row k)

typedef float v2f __attribute__((ext_vector_type(2)));
typedef float v8f __attribute__((ext_vector_type(8)));

// Force the native GLOBAL_ATOMIC_ADD_F32 (no CAS loop, no return -> STOREcnt).
__device__ __forceinline__ void atomic_add_f32(float* p, float v) {
    unsafeAtomicAdd(p, v);
}

// ---------------------------------------------------------------------------
// Edge scatter: one wave (32 lanes) per edge.
//   agg[dst] += x[src] * w   (float4 per lane, 4 hardware f32 atomics)
//   denom[dst] += w          (lane 0)
// ---------------------------------------------------------------------------
__global__ __launch_bounds__(256) void scatter_kernel(
    const float* __restrict__ x,
    const int*   __restrict__ src_idx,
    const int*   __restrict__ dst_idx,
    const float* __restrict__ w,
    float* __restrict__ agg,
    float* __restrict__ denom,
    int E)
{
    size_t gid = (size_t)blockIdx.x * blockDim.x + threadIdx.x;
    int e    = (int)(gid >> 5);
    int lane = (int)(gid & 31);
    if (e >= E) return;
    int s = src_idx[e];
    int d = dst_idx[e];
    float we = w[e];
    const float4 xs = *(const float4*)(x + (size_t)s * DIM + lane * 4);
    float* dp = agg + (size_t)d * DIM + lane * 4;
    atomic_add_f32(dp + 0, xs.x * we);
    atomic_add_f32(dp + 1, xs.y * we);
    atomic_add_f32(dp + 2, xs.z * we);
    atomic_add_f32(dp + 3, xs.w * we);
    if (lane == 0) atomic_add_f32(denom + d, we);
}

// ---------------------------------------------------------------------------
// Fused SAGE layer (in-place on x):
//   x = relu( x @ W_self + (agg/max(denom,1e-12)) @ W_neigh + bias )
// One 256-thread block per 16-row tile; wave w handles output cols [16w,16w+16).
// Weights staged in padded LDS (147 KB). WMMA: V_WMMA_F32_16X16X4_F32, two
// independent accumulator chains over a K=128 loop each.
// ---------------------------------------------------------------------------
__global__ __launch_bounds__(256) void sage_gemm_kernel(
    float*       __restrict__ x,      // [N, 128] in/out
    const float* __restrict__ agg,    // [N, 128]
    const float* __restrict__ denom,  // [N]
    const float* __restrict__ Wself,  // [128, 128]
    const float* __restrict__ Wneigh, // [128, 128]
    const float* __restrict__ bias,   // [128]
    int N)
{
    // padded rows: 0..127 = W_self, 128..255 = W_neigh
    __shared__ float lw[2 * DIM * LDS_STRIDE];   // 147456 B (<= 320 KB/WGP)

    const int t = threadIdx.x;
    {   // stage: thread t copies combined row t (128 floats) via float4
        const float* srow = (t < DIM) ? (Wself + (size_t)t * DIM)
                                      : (Wneigh + (size_t)(t - DIM) * DIM);
        float* drow = lw + (size_t)t * LDS_STRIDE;
        #pragma unroll
        for (int j = 0; j < DIM; j += 4)
            *(float4*)(drow + j) = *(const float4*)(srow + j);
    }
    __syncthreads();

    const int wave = t >> 5;          // 0..7 -> column tile
    const int lane = t & 31;
    const int half = lane >> 4;       // 0 | 1
    const int n    = lane & 15;       // col within tile / row within A tile
    const int koff = half * 2;        // A/B K sub-offset per ISA layout
    const int row0 = blockIdx.x * 16;
    const int col0 = wave * 16;

    int arow = row0 + n;
    if (arow > N - 1) arow = N - 1;   // safe clamp (N=50000 is exact multiple)

    const float dv = denom[arow];
    const float rscale = 1.0f / fmaxf(dv, 1e-12f);

    const float* xrow   = x   + (size_t)arow * DIM;
    const float* arow_p = agg + (size_t)arow * DIM;

    const float* lwb = lw + col0 + n;                 // B base for this lane

    v8f c0 = {};   // x @ W_self
    v8f c1 = {};   // agg_norm @ W_neigh

    #pragma unroll 4
    for (int k0 = 0; k0 < DIM; k0 += 4) {
        // A fragments: float2 at (k0+koff), consecutive K pair per lane half
        v2f a0 = *(const v2f*)(xrow + k0 + koff);
        v2f at = *(const v2f*)(arow_p + k0 + koff);
        v2f a1; a1[0] = at[0] * rscale; a1[1] = at[1] * rscale;

        // B fragments from LDS: VGPR0 -> row k0+koff, VGPR1 -> row k0+koff+1
        v2f b0, b1;
        b0[0] = lwb[(k0 + koff    ) * LDS_STRIDE];
        b0[1] = lwb[(k0 + koff + 1) * LDS_STRIDE];
        b1[0] = lwb[(DIM + k0 + koff    ) * LDS_STRIDE];
        b1[1] = lwb[(DIM + k0 + koff + 1) * LDS_STRIDE];

        c0 = __builtin_amdgcn_wmma_f32_16x16x4_f32(
                 false, a0, false, b0, (short)0, c0, false, false);
        c1 = __builtin_amdgcn_wmma_f32_16x16x4_f32(
                 false, a1, false, b1, (short)0, c1, false, false);
    }

    const float bv = bias[col0 + n];

    // all reads of this block's x rows are done before anyone overwrites them
    __syncthreads();

    // C/D layout: VGPR r -> row (r + half*8), col = col0 + n
    float* orow_base = x + (size_t)(row0 + half * 8) * DIM + col0 + n;
    if (row0 + 16 <= N) {             // uniform fast path: no per-row branches
        #pragma unroll
        for (int r = 0; r < 8; ++r) {
            float v = c0[r] + c1[r] + bv;
            orow_base[(size_t)r * DIM] = fmaxf(v, 0.0f);
        }
    } else {
        #pragma unroll
        for (int r = 0; r < 8; ++r) {
            int orow = row0 + r + half * 8;
            if (orow < N) {
                float v = c0[r] + c1[r] + bv;
                x[(size_t)orow * DIM + col0 + n] = fmaxf(v, 0.0f);
            }
        }
    }
}

// ---------------------------------------------------------------------------
// Final gather: out[q] = x[x_nodes[q]] + effect_emb[effect_ids[q]]
// One wave per query, float4 per lane.
// ---------------------------------------------------------------------------
__global__ __launch_bounds__(256) void gather_kernel(
    const float* __restrict__ x,
    const int*   __restrict__ nodes,
    const int*   __restrict__ effs,
    const float* __restrict__ emb,
    float* __restrict__ out,
    int Q)
{
    size_t gid = (size_t)blockIdx.x * blockDim.x + threadIdx.x;
    int q    = (int)(gid >> 5);
    int lane = (int)(gid & 31);
    if (q >= Q) return;
    int nd = nodes[q];
    int ef = effs[q];
    const float4 a = *(const float4*)(x   + (size_t)nd * DIM + lane * 4);
    const float4 b = *(const float4*)(emb + (size_t)ef * DIM + lane * 4);
    float4 r;
    r.x = a.x + b.x; r.y = a.y + b.y; r.z = a.z + b.z; r.w = a.w + b.w;
    *(float4*)(out + (size_t)q * DIM + lane * 4) = r;
}

// ---------------------------------------------------------------------------
extern "C" void kernel_launch(void* const* d_in, const int* in_sizes, int n_in,
                              void* d_out, int out_size, void* d_ws, size_t ws_size,
                              hipStream_t stream) {
    const float* graph_x    = (const float*)d_in[0];   // [N,128]
    const int*   edge_index = (const int*)  d_in[1];   // [2,E]
    const int*   x_nodes    = (const int*)  d_in[2];   // [Q]
    const int*   effect_ids = (const int*)  d_in[3];   // [Q]
    const float* chem_sim   = (const float*)d_in[4];   // [E]
    const float* W_self     = (const float*)d_in[5];   // [2,128,128]
    const float* W_neigh    = (const float*)d_in[6];   // [2,128,128]
    const float* bias       = (const float*)d_in[7];   // [2,128]
    const float* effect_emb = (const float*)d_in[8];   // [NEFF,128]
    float*       out        = (float*)d_out;           // [Q,128]

    const int N = in_sizes[0] / DIM;
    const int E = in_sizes[4];
    const int Q = in_sizes[2];

    // workspace layout: xbuf [N*128] | agg [N*128] | denom [N]
    float* xbuf  = (float*)d_ws;
    float* agg   = xbuf + (size_t)N * DIM;
    float* denom = agg  + (size_t)N * DIM;

    hipMemcpyAsync(xbuf, graph_x, (size_t)N * DIM * sizeof(float),
                   hipMemcpyDeviceToDevice, stream);

    const int HOPS = 2;
    for (int h = 0; h < HOPS; ++h) {
        // zero agg + denom (contiguous) — capturable async memset
        hipMemsetAsync(agg, 0, ((size_t)N * DIM + (size_t)N) * sizeof(float), stream);

        const size_t sthreads = (size_t)E * 32;
        scatter_kernel<<<(int)((sthreads + 255) / 256), 256, 0, stream>>>(
            xbuf, edge_index, edge_index + E, chem_sim, agg, denom, E);

        sage_gemm_kernel<<<(N + 15) / 16, 256, 0, stream>>>(
            xbuf, agg, denom,
            W_self  + (size_t)h * DIM * DIM,
            W_neigh + (size_t)h * DIM * DIM,
            bias    + (size_t)h * DIM,
            N);
    }

    const size_t gthreads = (size_t)Q * 32;
    gather_kernel<<<(int)((gthreads + 255) / 256), 256, 0, stream>>>(
        xbuf, x_nodes, effect_ids, effect_emb, out, Q);
}